// UnbatchedHMMTrajNet_21612275433733
// MI455X (gfx1250) — compile-verified
//
#include <hip/hip_runtime.h>
#include <stdint.h>

// ---------------------------------------------------------------------------
// HMM forward (rank-1 + diagonal transition), T=8192, B=128, A=128.
//   f_new[j] = al[j] + logaddexp(S + start[j], f[j] + cont[j]),
//   S = logsumexp_i(f[i] + beta[i])
// Exp-space with per-step renormalization: F = exp(f - c),
//   s   = sum_i F[i]*Beta[i]
//   F'  = ALexp[j] * (Start[j] + F[j]*Cont[j]/s),   c += log(s)
// Serial latency problem -> single wave32, 4 states/lane, shuffle reductions,
// deep async global->LDS prefetch pipeline (CDNA5 ASYNCcnt path) backed by
// speculative global_prefetch into L2 (input = 16 MB << 192 MB L2).
// ---------------------------------------------------------------------------

#define DEPTH 8      // async ring depth (7 instr/step, <=56 outstanding, cnt max 63)
#define PF_DIST 32   // global_prefetch distance (steps) to stage HBM -> L2

#define ASYNC_B128(ldsoff, gaddr) \
  asm volatile("global_load_async_to_lds_b128 %0, %1, off" :: "v"(ldsoff), "v"(gaddr) : "memory")
#define ASYNC_B32(ldsoff, gaddr) \
  asm volatile("global_load_async_to_lds_b32 %0, %1, off" :: "v"(ldsoff), "v"(gaddr) : "memory")
#define S_WAIT_ASYNC(n) asm volatile("s_wait_asynccnt " #n ::: "memory")
#define S_WAIT_DS0()    asm volatile("s_wait_dscnt 0" ::: "memory")

struct Smem {
  // per-step slot (2048 B): floats [0..255]=stop pairs, [256..383]=start, [384..511]=al
  float ring[DEPTH][512];
  int   actions[8192];
};

// low 32 bits of a flat pointer into LDS == wave-relative LDS byte address
__device__ __forceinline__ uint32_t lds_addr(const void* p) {
  return (uint32_t)(uintptr_t)p;
}

__device__ __forceinline__ float wave_sum(float v) {
  v += __shfl_xor(v, 16, 32);
  v += __shfl_xor(v, 8, 32);
  v += __shfl_xor(v, 4, 32);
  v += __shfl_xor(v, 2, 32);
  v += __shfl_xor(v, 1, 32);
  return v;
}

struct Pipe { float Beta[4], Cont[4], Sx[4], Ax[4]; };

__device__ __forceinline__ void load_exp(const float* slot, int lane, Pipe& P) {
#pragma unroll
  for (int j = 0; j < 4; ++j) {
    int b = lane + 32 * j;
    P.Beta[j] = __expf(slot[2 * b]);       // exp(stop_logp[b, STOP])
    P.Cont[j] = __expf(slot[2 * b + 1]);   // exp(stop_logp[b, CONT])
    P.Sx[j]   = __expf(slot[256 + b]);     // exp(start_logp[b])
    P.Ax[j]   = __expf(slot[384 + b]);     // exp(al[b])
  }
}

// Issue the 7 async DMA loads staging step u into its ring slot.
// s_wait_dscnt 0 first: guarantees every earlier ds_load of the slot being
// overwritten has completed (formal WAR closure; effectively free here).
__device__ __forceinline__ void issue_step(int u, int lane, Smem* sm,
                                           const float* stop, const float* start,
                                           const float* al, int A) {
  int a = sm->actions[u];                               // LDS broadcast read
  S_WAIT_DS0();
  uint32_t slot = lds_addr(&sm->ring[u & (DEPTH - 1)][0]);
  const float* stp = stop + (size_t)u * 256;
  ASYNC_B128((uint32_t)(slot + lane * 16),
             (unsigned long long)(uintptr_t)(stp + lane * 4));
  ASYNC_B128((uint32_t)(slot + 512 + lane * 16),
             (unsigned long long)(uintptr_t)(stp + 128 + lane * 4));
  ASYNC_B128((uint32_t)(slot + 1024 + lane * 16),
             (unsigned long long)(uintptr_t)(start + (size_t)u * 128 + lane * 4));
  const float* ap = al + ((size_t)u * 128) * (size_t)A + a;   // gather column actions[u]
#pragma unroll
  for (int j = 0; j < 4; ++j) {
    int b = lane + 32 * j;
    ASYNC_B32((uint32_t)(slot + 1536 + b * 4),
              (unsigned long long)(uintptr_t)(ap + (size_t)b * (size_t)A));
  }
}

// Speculative prefetch of step u's working set into near caches / L2.
__device__ __forceinline__ void prefetch_step(int u, int lane, Smem* sm,
                                              const float* stop, const float* start,
                                              const float* al, int A) {
  int a = sm->actions[u];
  const char* stp = (const char*)(stop + (size_t)u * 256);
  __builtin_prefetch(stp + lane * 64, 0, 3);                       // stop row (1 KB+)
  __builtin_prefetch((const char*)(start + (size_t)u * 128) + lane * 16, 0, 3);
  const float* ap = al + ((size_t)u * 128) * (size_t)A + a;
#pragma unroll
  for (int j = 0; j < 4; ++j)                                      // 128 gather lines
    __builtin_prefetch(ap + (size_t)(lane + 32 * j) * (size_t)A, 0, 3);
}

__global__ void __launch_bounds__(32)
hmm_forward_kernel(const float* __restrict__ action_logps,
                   const float* __restrict__ stop_logps,
                   const float* __restrict__ start_logps,
                   const int*   __restrict__ actions,
                   float* __restrict__ out, int T, int A) {
  __shared__ Smem sm;
  const int lane = threadIdx.x & 31;

  // Stage actions[] into LDS via async DMA (512 B / instruction).
  for (int i = 0; i < T; i += 128) {
    ASYNC_B128(lds_addr(&sm.actions[i]) + (uint32_t)(lane * 16),
               (unsigned long long)(uintptr_t)(actions + i) + (unsigned long long)(lane * 16));
  }
  S_WAIT_ASYNC(0);

  // Prologue: fill the ring for steps 0..DEPTH-1, pre-warm L2 for the next PF_DIST.
  for (int u = 0; u < DEPTH; ++u)
    issue_step(u, lane, &sm, stop_logps, start_logps, action_logps, A);
  for (int u = DEPTH; u < DEPTH + PF_DIST && u <= T - 1; ++u)
    prefetch_step(u, lane, &sm, stop_logps, start_logps, action_logps, A);

  S_WAIT_ASYNC(42);   // steps 0 and 1 resident (7*(DEPTH-2))

  // f0 = start[0] + al[0]  ->  F = exp(f0), c = 0
  float F[4];
  {
    const float* s0 = &sm.ring[0][0];
#pragma unroll
    for (int j = 0; j < 4; ++j) {
      int b = lane + 32 * j;
      F[j] = __expf(s0[256 + b] + s0[384 + b]);
    }
  }
  Pipe P;
  load_exp(&sm.ring[1 & (DEPTH - 1)][0], lane, P);

  float c = 0.0f;
  for (int t = 1; t <= T - 1; ++t) {
    // s = sum_i F[i] * Beta[i]  (the rank-1 coupling term)
    float sp = 0.0f;
#pragma unroll
    for (int j = 0; j < 4; ++j) sp = fmaf(F[j], P.Beta[j], sp);
    float s   = wave_sum(sp);
    float inv = __builtin_amdgcn_rcpf(s);
    c += __logf(s);
#pragma unroll
    for (int j = 0; j < 4; ++j)
      F[j] = P.Ax[j] * fmaf(F[j] * P.Cont[j], inv, P.Sx[j]);

    if (t + 1 <= T - 1) {
      S_WAIT_ASYNC(42);                                   // slot t+1 resident
      load_exp(&sm.ring[(t + 1) & (DEPTH - 1)][0], lane, P);
    }
    if (t + DEPTH <= T - 1)
      issue_step(t + DEPTH, lane, &sm, stop_logps, start_logps, action_logps, A);
    if (t + DEPTH + PF_DIST <= T - 1)
      prefetch_step(t + DEPTH + PF_DIST, lane, &sm,
                    stop_logps, start_logps, action_logps, A);
  }

  // total = logsumexp_j(f[j] + stop_logps[T, j, STOP]);  out = -total
  float fp = 0.0f;
#pragma unroll
  for (int j = 0; j < 4; ++j) {
    int b = lane + 32 * j;
    fp = fmaf(F[j], __expf(stop_logps[(size_t)T * 256 + 2 * b]), fp);
  }
  float tot = wave_sum(fp);
  if (lane == 0) out[0] = -(c + __logf(tot));
}

extern "C" void kernel_launch(void* const* d_in, const int* in_sizes, int n_in,
                              void* d_out, int out_size, void* d_ws, size_t ws_size,
                              hipStream_t stream) {
  (void)n_in; (void)out_size; (void)d_ws; (void)ws_size;
  const float* action_logps = (const float*)d_in[0];
  const float* stop_logps   = (const float*)d_in[1];
  const float* start_logps  = (const float*)d_in[2];
  const int*   actions      = (const int*)d_in[3];
  float* out = (float*)d_out;

  const int T = in_sizes[3];                       // 8192
  const int B = in_sizes[2] / (T + 1);             // 128 (kernel assumes 128)
  const int A = in_sizes[0] / ((T + 1) * B);       // 128

  hmm_forward_kernel<<<1, 32, 0, stream>>>(action_logps, stop_logps, start_logps,
                                           actions, out, T, A);
}